// GNNModule_44607530336764
// MI455X (gfx1250) — compile-verified
//
#include <hip/hip_runtime.h>

typedef __attribute__((ext_vector_type(2))) float v2f;
typedef __attribute__((ext_vector_type(4))) float v4f;
typedef __attribute__((ext_vector_type(8))) float v8f;

#define DIM 64
#define PSTRIDE 160  // dwords per K-pair row of staged weights (128 data + 32 pad -> halves hit disjoint bank sets)
#define TSTRIDE 66   // floats per row of per-wave tile (even -> 8B-aligned b64 A-fragment loads)
#define WAVES 4      // waves per block in the MLP kernel

// ---------------------------------------------------------------------------
// Kernel 1: edge scatter  agg[dst] += relu(x[src] + edge_attr)
// 16 lanes per edge, float4 per lane. edge_attr / indices are streamed once ->
// nontemporal (keep x & agg resident in the 192MB L2). x row = one 256B
// segment shared by 16 lanes. Atomics: no-return global_atomic_add_f32.
// ---------------------------------------------------------------------------
__global__ __launch_bounds__(256) void gine_scatter(
    const float* __restrict__ x,
    const int* __restrict__ src,
    const int* __restrict__ dst,
    const float* __restrict__ ea,
    float* __restrict__ agg,
    int E)
{
    long long gid = (long long)blockIdx.x * blockDim.x + threadIdx.x;
    int e = (int)(gid >> 4);
    if (e >= E) return;
    int c = ((int)gid & 15) << 2;

    int s = __builtin_nontemporal_load(src + e);
    int d = __builtin_nontemporal_load(dst + e);

    const v4f xv = *(const v4f*)(x + (size_t)s * DIM + c);
    const v4f ev = __builtin_nontemporal_load((const v4f*)(ea + (size_t)e * DIM + c));

    float m0 = fmaxf(xv.x + ev.x, 0.0f);
    float m1 = fmaxf(xv.y + ev.y, 0.0f);
    float m2 = fmaxf(xv.z + ev.z, 0.0f);
    float m3 = fmaxf(xv.w + ev.w, 0.0f);

    float* ap = agg + (size_t)d * DIM + c;
#if defined(__HIP_DEVICE_COMPILE__)
    unsafeAtomicAdd(ap + 0, m0);   // -> global_atomic_add_f32 (no return)
    unsafeAtomicAdd(ap + 1, m1);
    unsafeAtomicAdd(ap + 2, m2);
    unsafeAtomicAdd(ap + 3, m3);
#else
    atomicAdd(ap + 0, m0); atomicAdd(ap + 1, m1);
    atomicAdd(ap + 2, m2); atomicAdd(ap + 3, m3);
#endif
}

// ---------------------------------------------------------------------------
// Kernel 2: fused last-layer GINE MLP using V_WMMA_F32_16X16X4_F32.
// out = relu( relu(((1+eps)*x + agg) @ W1 + b1) @ W2 + b2 )
//
// Fragment layouts (per ISA):
//   A 16x4 f32 : lane L holds M = L%16, K = kb + {0,1}, kb = s*4 + (L/16)*2
//   B 4x16 f32 : lane L holds N = L%16, same K pair
//   C/D 16x16  : VGPR r holds M = r + 8*(L/16), N = L%16
// Weights staged K-pair-major so each B fragment is ONE ds_load_b64:
//   sW[(k/2)*PSTRIDE + 2*n + (k&1)] = W[k][n]
// ---------------------------------------------------------------------------
__global__ __launch_bounds__(WAVES * 32) void gine_mlp(
    const float* __restrict__ x,
    const float* __restrict__ agg,
    const float* __restrict__ W1,
    const float* __restrict__ b1,
    const float* __restrict__ W2,
    const float* __restrict__ b2,
    const float* __restrict__ epsArr,
    int lastL,
    float* __restrict__ out,
    int N)
{
    __shared__ float sW1[(DIM / 2) * PSTRIDE];
    __shared__ float sW2[(DIM / 2) * PSTRIDE];
    __shared__ float sB1[DIM];
    __shared__ float sB2[DIM];
    __shared__ float sT[WAVES][16 * TSTRIDE];

    const int tid    = threadIdx.x;
    const int wave   = tid >> 5;
    const int lane   = tid & 31;
    const int half   = lane >> 4;   // 0 or 1
    const int lanelo = lane & 15;

    // Stage last-layer weights (K-pair-major) and biases in LDS.
    const float* W1l = W1 + (size_t)lastL * DIM * DIM;
    const float* W2l = W2 + (size_t)lastL * DIM * DIM;
    for (int i = tid; i < DIM * DIM; i += blockDim.x) {
        int k = i >> 6, n = i & 63;
        int p = (k >> 1) * PSTRIDE + 2 * n + (k & 1);
        sW1[p] = W1l[i];
        sW2[p] = W2l[i];
    }
    for (int i = tid; i < DIM; i += blockDim.x) {
        sB1[i] = b1[lastL * DIM + i];
        sB2[i] = b2[lastL * DIM + i];
    }
    __syncthreads();

    const float eps1    = 1.0f + epsArr[lastL];
    const int   rowBase = (blockIdx.x * WAVES + wave) * 16;
    if (rowBase >= N) return;          // wave-uniform: EXEC stays all-1s

    float* tile = &sT[wave][0];

    // h0 tile = eps1 * x + agg  (coalesced 128b global loads, b64 LDS stores)
    for (int i = 0; i < 8; ++i) {
        int f   = i * 32 + lane;       // float4 index within 16x64 tile
        int row = f >> 4;
        int col = (f & 15) << 2;
        const v4f xv = *(const v4f*)(x   + (size_t)(rowBase + row) * DIM + col);
        const v4f av = *(const v4f*)(agg + (size_t)(rowBase + row) * DIM + col);
        v2f h01, h23;
        h01.x = fmaf(eps1, xv.x, av.x);
        h01.y = fmaf(eps1, xv.y, av.y);
        h23.x = fmaf(eps1, xv.z, av.z);
        h23.y = fmaf(eps1, xv.w, av.w);
        *(v2f*)(tile + row * TSTRIDE + col)     = h01;   // 8B aligned (TSTRIDE even)
        *(v2f*)(tile + row * TSTRIDE + col + 2) = h23;
    }

    v8f acc0, acc1, acc2, acc3;

    // ---------------- GEMM1: t = relu(h0 @ W1 + b1) ----------------
    {
        float bb0 = sB1[ 0 + lanelo];
        float bb1 = sB1[16 + lanelo];
        float bb2 = sB1[32 + lanelo];
        float bb3 = sB1[48 + lanelo];
        for (int r = 0; r < 8; ++r) {
            acc0[r] = bb0; acc1[r] = bb1; acc2[r] = bb2; acc3[r] = bb3;
        }
    }
    for (int s = 0; s < 16; ++s) {
        const int kb = s * 4 + half * 2;           // even
        const int k2 = s * 2 + half;               // K-pair row
        v2f a = *(const v2f*)(tile + lanelo * TSTRIDE + kb);
        v2f bf0 = *(const v2f*)(sW1 + k2 * PSTRIDE + 2 * ( 0 + lanelo));
        v2f bf1 = *(const v2f*)(sW1 + k2 * PSTRIDE + 2 * (16 + lanelo));
        v2f bf2 = *(const v2f*)(sW1 + k2 * PSTRIDE + 2 * (32 + lanelo));
        v2f bf3 = *(const v2f*)(sW1 + k2 * PSTRIDE + 2 * (48 + lanelo));
        acc0 = __builtin_amdgcn_wmma_f32_16x16x4_f32(false, a, false, bf0, (short)0, acc0, false, false);
        acc1 = __builtin_amdgcn_wmma_f32_16x16x4_f32(false, a, false, bf1, (short)0, acc1, false, false);
        acc2 = __builtin_amdgcn_wmma_f32_16x16x4_f32(false, a, false, bf2, (short)0, acc2, false, false);
        acc3 = __builtin_amdgcn_wmma_f32_16x16x4_f32(false, a, false, bf3, (short)0, acc3, false, false);
    }

    // ReLU + transpose through own-wave LDS tile (DScnt keeps order in-wave).
    for (int r = 0; r < 8; ++r) {
        int m = r + half * 8;
        tile[m * TSTRIDE +  0 + lanelo] = fmaxf(acc0[r], 0.0f);
        tile[m * TSTRIDE + 16 + lanelo] = fmaxf(acc1[r], 0.0f);
        tile[m * TSTRIDE + 32 + lanelo] = fmaxf(acc2[r], 0.0f);
        tile[m * TSTRIDE + 48 + lanelo] = fmaxf(acc3[r], 0.0f);
    }

    // ---------------- GEMM2: out = relu(t @ W2 + b2) ----------------
    {
        float bb0 = sB2[ 0 + lanelo];
        float bb1 = sB2[16 + lanelo];
        float bb2 = sB2[32 + lanelo];
        float bb3 = sB2[48 + lanelo];
        for (int r = 0; r < 8; ++r) {
            acc0[r] = bb0; acc1[r] = bb1; acc2[r] = bb2; acc3[r] = bb3;
        }
    }
    for (int s = 0; s < 16; ++s) {
        const int kb = s * 4 + half * 2;
        const int k2 = s * 2 + half;
        v2f a = *(const v2f*)(tile + lanelo * TSTRIDE + kb);
        v2f bf0 = *(const v2f*)(sW2 + k2 * PSTRIDE + 2 * ( 0 + lanelo));
        v2f bf1 = *(const v2f*)(sW2 + k2 * PSTRIDE + 2 * (16 + lanelo));
        v2f bf2 = *(const v2f*)(sW2 + k2 * PSTRIDE + 2 * (32 + lanelo));
        v2f bf3 = *(const v2f*)(sW2 + k2 * PSTRIDE + 2 * (48 + lanelo));
        acc0 = __builtin_amdgcn_wmma_f32_16x16x4_f32(false, a, false, bf0, (short)0, acc0, false, false);
        acc1 = __builtin_amdgcn_wmma_f32_16x16x4_f32(false, a, false, bf1, (short)0, acc1, false, false);
        acc2 = __builtin_amdgcn_wmma_f32_16x16x4_f32(false, a, false, bf2, (short)0, acc2, false, false);
        acc3 = __builtin_amdgcn_wmma_f32_16x16x4_f32(false, a, false, bf3, (short)0, acc3, false, false);
    }

    // Final ReLU + nontemporal store (out is write-once).
    for (int r = 0; r < 8; ++r) {
        int m = rowBase + r + half * 8;
        __builtin_nontemporal_store(fmaxf(acc0[r], 0.0f), out + (size_t)m * DIM +  0 + lanelo);
        __builtin_nontemporal_store(fmaxf(acc1[r], 0.0f), out + (size_t)m * DIM + 16 + lanelo);
        __builtin_nontemporal_store(fmaxf(acc2[r], 0.0f), out + (size_t)m * DIM + 32 + lanelo);
        __builtin_nontemporal_store(fmaxf(acc3[r], 0.0f), out + (size_t)m * DIM + 48 + lanelo);
    }
}

// ---------------------------------------------------------------------------
// Launch: inputs are (x, edge_index, edge_attr, W1, b1, W2, b2, eps).
// Only the LAST layer's MLP matters (reference re-reads original x per layer,
// so every iteration's `out` is overwritten; agg is layer-invariant).
// ---------------------------------------------------------------------------
extern "C" void kernel_launch(void* const* d_in, const int* in_sizes, int n_in,
                              void* d_out, int out_size, void* d_ws, size_t ws_size,
                              hipStream_t stream)
{
    const float* x   = (const float*)d_in[0];
    const int*   ei  = (const int*)  d_in[1];   // [2, E] row-major
    const float* ea  = (const float*)d_in[2];
    const float* W1  = (const float*)d_in[3];
    const float* b1  = (const float*)d_in[4];
    const float* W2  = (const float*)d_in[5];
    const float* b2  = (const float*)d_in[6];
    const float* eps = (const float*)d_in[7];

    const int N = in_sizes[0] / DIM;
    const int E = in_sizes[2] / DIM;
    const int L = in_sizes[7];       // number of layers

    float* agg = (float*)d_ws;
    (void)hipMemsetAsync(agg, 0, (size_t)N * DIM * sizeof(float), stream);

    const int* src = ei;
    const int* dst = ei + E;

    {
        long long total = (long long)E * 16;
        int blocks = (int)((total + 255) / 256);
        gine_scatter<<<blocks, 256, 0, stream>>>(x, src, dst, ea, agg, E);
    }
    {
        int tiles  = (N + 15) / 16;
        int blocks = (tiles + WAVES - 1) / WAVES;
        gine_mlp<<<blocks, WAVES * 32, 0, stream>>>(x, agg, W1, b1, W2, b2,
                                                    eps, L - 1, (float*)d_out, N);
    }
}